// Network_74320114090270
// MI455X (gfx1250) — compile-verified
//
#include <hip/hip_runtime.h>

typedef __attribute__((ext_vector_type(2))) float v2f;
typedef __attribute__((ext_vector_type(8))) float v8f;

#define BATCH    32
#define NPTS_ENC 2048
#define NPTS_DEC 2116          // 46*46
#define M_ENC    (BATCH * NPTS_ENC)   // 65536
#define M_DEC    (BATCH * NPTS_DEC)   // 67712
#define N_SAMPLE 2048

// ---------------------------------------------------------------------------
// Fold BN params: scale = g * rsqrt(v+eps); bias = scale*(b_lin - m) + b_bn
// ---------------------------------------------------------------------------
__global__ void fold_bn_kernel(const float* __restrict__ blin,
                               const float* __restrict__ g,
                               const float* __restrict__ bb,
                               const float* __restrict__ m,
                               const float* __restrict__ v,
                               float* __restrict__ scale,
                               float* __restrict__ bias, int O) {
    int o = blockIdx.x * blockDim.x + threadIdx.x;
    if (o < O) {
        float s = g[o] * rsqrtf(v[o] + 1e-5f);
        scale[o] = s;
        bias[o]  = s * (blin[o] - m[o]) + bb[o];
    }
}

// ---------------------------------------------------------------------------
// Encoder conv1 (K=3 too small for WMMA): H1[m,o] = relu(s*W[o,:]·x[m] + b)
// ---------------------------------------------------------------------------
__global__ __launch_bounds__(256) void enc_c1_kernel(
    const float* __restrict__ x, const float* __restrict__ W,
    const float* __restrict__ scale, const float* __restrict__ bias,
    float* __restrict__ H1) {
    int t = blockIdx.x * 256 + threadIdx.x;     // < M_ENC*64
    int m = t >> 6, o = t & 63;
    const float* xp = x + (size_t)m * 3;
    float a = W[o * 3 + 0] * xp[0] + W[o * 3 + 1] * xp[1] + W[o * 3 + 2] * xp[2];
    a = a * scale[o] + bias[o];
    H1[t] = fmaxf(a, 0.0f);
}

// ---------------------------------------------------------------------------
// fp32 WMMA GEMM: OUT[M,O] = act(scale[o]*(A[M,K] @ W[O,K]^T) + bias[o])
// Block = 256 thr = 8 waves; block tile 64(M) x 128(N); each wave owns a
// 2x2 grid of 16x16 accumulators (4 WMMAs per 2 A-frags + 2 B-frags).
// K staged through LDS 32-deep; both tiles stored with stride 36 floats
// (16B aligned rows, conflict-free 32-lane fragment reads).
// Staging uses gfx1250 GLOBAL_LOAD_ASYNC_TO_LDS_B128 when available.
// Requires M%64==0, O%128==0, K%32==0 (true for all layers used).
// ---------------------------------------------------------------------------
#define AS_LD 36
#define WS_LD 36

#if defined(__has_builtin)
#if __has_builtin(__builtin_amdgcn_global_load_async_to_lds_b128) && \
    __has_builtin(__builtin_amdgcn_s_wait_asynccnt)
#define USE_ASYNC_LDS 1
#endif
#endif

#ifdef USE_ASYNC_LDS
// builtin signature (from clang diagnostic): param0 = vector_size(16) int in
// AS1 (printed as __device__), so use matching int4-vector pointee types.
typedef int b128_t __attribute__((vector_size(4 * sizeof(int))));
typedef __attribute__((address_space(1))) b128_t* glb128_t;
typedef __attribute__((address_space(3))) b128_t* lds128_t;
#define ASYNC_B128(gsrc, ldst) \
    __builtin_amdgcn_global_load_async_to_lds_b128((glb128_t)(gsrc), (lds128_t)(ldst), 0, 0)
#endif

__global__ __launch_bounds__(256) void gemm_wmma_kernel(
    const float* __restrict__ A, const float* __restrict__ W,
    const float* __restrict__ scale, const float* __restrict__ bias,
    float* __restrict__ OUT, int M, int K, int O, int relu) {
    __shared__ float As[64 * AS_LD];    // 9 KB
    __shared__ float Ws[128 * WS_LD];   // 18 KB  (stored [n][k])

    const int tid  = threadIdx.x;
    const int lane = tid & 31;
    const int wave = tid >> 5;
    const int rbase = (wave >> 2) * 32;        // 0 / 32
    const int cbase = (wave & 3) * 32;         // 0 / 32 / 64 / 96
    const int mBase = blockIdx.x * 64;
    const int nBase = blockIdx.y * 128;
    const int l15 = lane & 15;
    const int hi  = (lane >> 4) << 1;          // lanes 0-15: K+0/+1, 16-31: K+2/+3

    v8f acc00 = {}, acc01 = {}, acc10 = {}, acc11 = {};

    for (int k0 = 0; k0 < K; k0 += 32) {
        // ---- stage A 64x32 (512 16B chunks, 2/thread) and W 128x32 (4/thread)
#ifdef USE_ASYNC_LDS
#pragma unroll
        for (int j = 0; j < 2; ++j) {
            int c = tid + j * 256; int row = c >> 3; int col = (c & 7) << 2;
            ASYNC_B128(&A[(size_t)(mBase + row) * K + k0 + col], &As[row * AS_LD + col]);
        }
#pragma unroll
        for (int j = 0; j < 4; ++j) {
            int c = tid + j * 256; int n = c >> 3; int k = (c & 7) << 2;
            ASYNC_B128(&W[(size_t)(nBase + n) * K + k0 + k], &Ws[n * WS_LD + k]);
        }
        __builtin_amdgcn_s_wait_asynccnt(0);
#else
#pragma unroll
        for (int j = 0; j < 2; ++j) {
            int c = tid + j * 256; int row = c >> 3; int col = (c & 7) << 2;
            *(float4*)&As[row * AS_LD + col] =
                *(const float4*)&A[(size_t)(mBase + row) * K + k0 + col];
        }
#pragma unroll
        for (int j = 0; j < 4; ++j) {
            int c = tid + j * 256; int n = c >> 3; int k = (c & 7) << 2;
            *(float4*)&Ws[n * WS_LD + k] =
                *(const float4*)&W[(size_t)(nBase + n) * K + k0 + k];
        }
#endif
        __syncthreads();

#pragma unroll
        for (int kk = 0; kk < 32; kk += 4) {
            v2f a0, a1, b0, b1;
            a0[0] = As[(rbase + l15) * AS_LD + kk + hi];
            a0[1] = As[(rbase + l15) * AS_LD + kk + hi + 1];
            a1[0] = As[(rbase + 16 + l15) * AS_LD + kk + hi];
            a1[1] = As[(rbase + 16 + l15) * AS_LD + kk + hi + 1];
            b0[0] = Ws[(cbase + l15) * WS_LD + kk + hi];
            b0[1] = Ws[(cbase + l15) * WS_LD + kk + hi + 1];
            b1[0] = Ws[(cbase + 16 + l15) * WS_LD + kk + hi];
            b1[1] = Ws[(cbase + 16 + l15) * WS_LD + kk + hi + 1];
            acc00 = __builtin_amdgcn_wmma_f32_16x16x4_f32(false, a0, false, b0, (short)0, acc00, false, false);
            acc01 = __builtin_amdgcn_wmma_f32_16x16x4_f32(false, a0, false, b1, (short)0, acc01, false, false);
            acc10 = __builtin_amdgcn_wmma_f32_16x16x4_f32(false, a1, false, b0, (short)0, acc10, false, false);
            acc11 = __builtin_amdgcn_wmma_f32_16x16x4_f32(false, a1, false, b1, (short)0, acc11, false, false);
        }
        __syncthreads();
    }

    // epilogue: C/D layout -> VGPR i: lanes0-15 row i, lanes16-31 row i+8
    const int rowHi = (lane >> 4) << 3;
    auto store_tile = [&](const v8f& acc, int rt, int ct) {
        int col = nBase + cbase + ct * 16 + l15;
        float sc = scale[col], bs = bias[col];
        int row0 = mBase + rbase + rt * 16 + rowHi;
#pragma unroll
        for (int i = 0; i < 8; ++i) {
            float val = acc[i] * sc + bs;
            if (relu) val = fmaxf(val, 0.0f);
            OUT[(size_t)(row0 + i) * O + col] = val;
        }
    };
    store_tile(acc00, 0, 0);
    store_tile(acc01, 0, 1);
    store_tile(acc10, 1, 0);
    store_tile(acc11, 1, 1);
}

// ---------------------------------------------------------------------------
// Global max-pool over points: feat[b,o] = max_p H3[b,p,o]
// ---------------------------------------------------------------------------
__global__ __launch_bounds__(256) void maxpool_kernel(
    const float* __restrict__ H3, float* __restrict__ feat) {
    int t = blockIdx.x * 256 + threadIdx.x;   // < 32*512
    int b = t >> 9, o = t & 511;
    const float* p = H3 + (size_t)b * NPTS_ENC * 512 + o;
    float mx = -3.4e38f;
    for (int i = 0; i < NPTS_ENC; ++i) mx = fmaxf(mx, p[(size_t)i * 512]);
    feat[t] = mx;
}

// ---------------------------------------------------------------------------
// base[b,o] = feat[b,:] . W[o, 0:512]  (feat part of concat layers f1/f4)
// ---------------------------------------------------------------------------
__global__ __launch_bounds__(256) void fold_base_kernel(
    const float* __restrict__ feat, const float* __restrict__ W, int ldw,
    float* __restrict__ base) {
    int t = blockIdx.x * 256 + threadIdx.x;   // < 32*512
    int b = t >> 9, o = t & 511;
    const float* f = feat + b * 512;
    const float* w = W + (size_t)o * ldw;
    float acc = 0.0f;
    for (int k = 0; k < 512; ++k) acc += f[k] * w[k];
    base[t] = acc;
}

// ---------------------------------------------------------------------------
// f1/f4 become elementwise: OUT[m,o]=relu(s*(base[b,o] + extra.Wext) + bias)
// extra==nullptr: use 46x46 folding grid (col0=lin[p%46], col1=lin[p/46])
// ---------------------------------------------------------------------------
__global__ __launch_bounds__(256) void fold_elem_kernel(
    const float* __restrict__ baseBO, const float* __restrict__ W, int ldw,
    const float* __restrict__ extra, const float* __restrict__ scale,
    const float* __restrict__ bias, float* __restrict__ OUT) {
    int t = blockIdx.x * 256 + threadIdx.x;   // < M_DEC*512
    int o = t & 511;
    int m = t >> 9;
    int b = m / NPTS_DEC;
    int p = m - b * NPTS_DEC;
    float acc = baseBO[b * 512 + o];
    const float* wext = W + (size_t)o * ldw + 512;
    if (extra) {
        const float* e = extra + (size_t)m * 3;
        acc += e[0] * wext[0] + e[1] * wext[1] + e[2] * wext[2];
    } else {
        float g0 = -0.5f + (float)(p % 46) * (1.0f / 45.0f);
        float g1 = -0.5f + (float)(p / 46) * (1.0f / 45.0f);
        acc += g0 * wext[0] + g1 * wext[1];
    }
    acc = acc * scale[o] + bias[o];
    OUT[t] = fmaxf(acc, 0.0f);
}

// ---------------------------------------------------------------------------
// Head f3/f6 (512 -> 3 + BN): one wave32 per point, lane-parallel dot + shuffle
// reduction (keeps reads of H fully coalesced).
// ---------------------------------------------------------------------------
__global__ __launch_bounds__(256) void head_kernel(
    const float* __restrict__ H, const float* __restrict__ W3,
    const float* __restrict__ scale, const float* __restrict__ bias,
    float* __restrict__ OUT, int Mtot) {
    int lane = threadIdx.x & 31;
    int gw = blockIdx.x * (blockDim.x >> 5) + (threadIdx.x >> 5);
    int nw = gridDim.x * (blockDim.x >> 5);
    for (int m = gw; m < Mtot; m += nw) {
        const float* h = H + (size_t)m * 512;
        float a0 = 0.f, a1 = 0.f, a2 = 0.f;
        for (int k = lane; k < 512; k += 32) {
            float vv = h[k];
            a0 += vv * W3[k];
            a1 += vv * W3[512 + k];
            a2 += vv * W3[1024 + k];
        }
        for (int off = 16; off; off >>= 1) {
            a0 += __shfl_down(a0, off);
            a1 += __shfl_down(a1, off);
            a2 += __shfl_down(a2, off);
        }
        if (lane == 0) {
            OUT[(size_t)m * 3 + 0] = a0 * scale[0] + bias[0];
            OUT[(size_t)m * 3 + 1] = a1 * scale[1] + bias[1];
            OUT[(size_t)m * 3 + 2] = a2 * scale[2] + bias[2];
        }
    }
}

// ---------------------------------------------------------------------------
// Farthest point sampling: one block per (batch, branch). Points + running
// min-dists LDS-resident; first sample is index 0; out[s] = pts[far_s].
// Tie-break = smallest index (matches jnp.argmax first-max semantics).
// ---------------------------------------------------------------------------
__global__ __launch_bounds__(256) void fps_kernel(
    const float* __restrict__ x1, const float* __restrict__ x2,
    float* __restrict__ out) {
    __shared__ float pts[NPTS_DEC * 3];
    __shared__ float dist[NPTS_DEC];
    __shared__ float rv[8];
    __shared__ int   ri[8];
    __shared__ int   farS;

    const int tid  = threadIdx.x;
    const int b    = blockIdx.x & 31;
    const int sel  = blockIdx.x >> 5;
    const float* src = (sel ? x2 : x1) + (size_t)b * NPTS_DEC * 3;
    float* dst = out + (size_t)sel * BATCH * N_SAMPLE * 3 + (size_t)b * N_SAMPLE * 3;

    for (int i = tid; i < NPTS_DEC * 3; i += 256) pts[i] = src[i];
    for (int i = tid; i < NPTS_DEC; i += 256) dist[i] = 1e10f;
    if (tid == 0) farS = 0;
    __syncthreads();

    const int lane = tid & 31, wave = tid >> 5;
    for (int s = 0; s < N_SAMPLE; ++s) {
        int far = farS;
        float cx = pts[far * 3], cy = pts[far * 3 + 1], cz = pts[far * 3 + 2];
        if (tid == 0) { dst[s * 3] = cx; dst[s * 3 + 1] = cy; dst[s * 3 + 2] = cz; }

        float best = -1.0f;
        int   bidx = NPTS_DEC;
        for (int i = tid; i < NPTS_DEC; i += 256) {
            float dx = pts[i * 3] - cx, dy = pts[i * 3 + 1] - cy, dz = pts[i * 3 + 2] - cz;
            float d  = dx * dx + dy * dy + dz * dz;
            float nd = fminf(dist[i], d);
            dist[i]  = nd;
            if (nd > best) { best = nd; bidx = i; }   // strict > keeps first max
        }
        for (int off = 16; off; off >>= 1) {
            float ov = __shfl_down(best, off);
            int   oi = __shfl_down(bidx, off);
            if (ov > best || (ov == best && oi < bidx)) { best = ov; bidx = oi; }
        }
        if (lane == 0) { rv[wave] = best; ri[wave] = bidx; }
        __syncthreads();
        if (tid == 0) {
            float bb = rv[0]; int bi = ri[0];
            for (int w = 1; w < 8; ++w)
                if (rv[w] > bb || (rv[w] == bb && ri[w] < bi)) { bb = rv[w]; bi = ri[w]; }
            farS = bi;
        }
        __syncthreads();
    }
}

// ---------------------------------------------------------------------------
// Host-side orchestration
// ---------------------------------------------------------------------------
extern "C" void kernel_launch(void* const* d_in, const int* in_sizes, int n_in,
                              void* d_out, int out_size, void* d_ws, size_t ws_size,
                              hipStream_t stream) {
    // setup_inputs() dict order: x, then params (lin W,b x9), then bn (g,b,m,v x9)
    const float* x    = (const float*)d_in[0];
    const float* c1W  = (const float*)d_in[1];   const float* c1b = (const float*)d_in[2];
    const float* c2W  = (const float*)d_in[3];   const float* c2b = (const float*)d_in[4];
    const float* c3W  = (const float*)d_in[5];   const float* c3b = (const float*)d_in[6];
    const float* f1W  = (const float*)d_in[7];   const float* f1b = (const float*)d_in[8];
    const float* f2W  = (const float*)d_in[9];   const float* f2b = (const float*)d_in[10];
    const float* f3W  = (const float*)d_in[11];  const float* f3b = (const float*)d_in[12];
    const float* f4W  = (const float*)d_in[13];  const float* f4b = (const float*)d_in[14];
    const float* f5W  = (const float*)d_in[15];  const float* f5b = (const float*)d_in[16];
    const float* f6W  = (const float*)d_in[17];  const float* f6b = (const float*)d_in[18];
    const float* const* bn = (const float* const*)(d_in + 19); // 9 x (g,b,m,v)

    // workspace layout (floats)
    float* ws = (float*)d_ws;
    const size_t SG = (size_t)M_DEC * 512;       // 34,668,544 (also holds H3)
    float* G1   = ws;                            // H3 / f1-out / f4-out
    float* G2   = G1 + SG;                       // f2-out / f5-out
    float* H1   = G2 + SG;                       // M_ENC*64
    float* H2   = H1 + (size_t)M_ENC * 64;       // M_ENC*128
    float* feat = H2 + (size_t)M_ENC * 128;      // 32*512
    float* base1 = feat + 32 * 512;
    float* base4 = base1 + 32 * 512;
    float* x1b  = base4 + 32 * 512;              // M_DEC*3
    float* x2b  = x1b + (size_t)M_DEC * 3;
    float* sb   = x2b + (size_t)M_DEC * 3;       // folded scale/bias
    const int Osz[9] = {64, 128, 512, 512, 512, 3, 512, 512, 3};
    float* sc[9]; float* bi[9];
    {
        float* p = sb;
        for (int i = 0; i < 9; ++i) { sc[i] = p; p += Osz[i]; bi[i] = p; p += Osz[i]; }
    }
    const float* linb[9] = {c1b, c2b, c3b, f1b, f2b, f3b, f4b, f5b, f6b};

    // 1) fold BN into per-channel scale/bias
    for (int i = 0; i < 9; ++i)
        fold_bn_kernel<<<(Osz[i] + 255) / 256, 256, 0, stream>>>(
            linb[i], bn[4 * i + 0], bn[4 * i + 1], bn[4 * i + 2], bn[4 * i + 3],
            sc[i], bi[i], Osz[i]);

    // 2) encoder
    enc_c1_kernel<<<(M_ENC * 64) / 256, 256, 0, stream>>>(x, c1W, sc[0], bi[0], H1);
    gemm_wmma_kernel<<<dim3(M_ENC / 64, 128 / 128), 256, 0, stream>>>(
        H1, c2W, sc[1], bi[1], H2, M_ENC, 64, 128, 1);
    gemm_wmma_kernel<<<dim3(M_ENC / 64, 512 / 128), 256, 0, stream>>>(
        H2, c3W, sc[2], bi[2], G1 /*H3*/, M_ENC, 128, 512, 0);
    maxpool_kernel<<<64, 256, 0, stream>>>(G1, feat);

    // 3) decoder fold 1
    fold_base_kernel<<<64, 256, 0, stream>>>(feat, f1W, 514, base1);
    fold_base_kernel<<<64, 256, 0, stream>>>(feat, f4W, 515, base4);
    fold_elem_kernel<<<((size_t)M_DEC * 512) / 256, 256, 0, stream>>>(
        base1, f1W, 514, nullptr, sc[3], bi[3], G1);
    gemm_wmma_kernel<<<dim3(M_DEC / 64, 512 / 128), 256, 0, stream>>>(
        G1, f2W, sc[4], bi[4], G2, M_DEC, 512, 512, 1);
    head_kernel<<<2048, 256, 0, stream>>>(G2, f3W, sc[5], bi[5], x1b, M_DEC);

    // 4) decoder fold 2
    fold_elem_kernel<<<((size_t)M_DEC * 512) / 256, 256, 0, stream>>>(
        base4, f4W, 515, x1b, sc[6], bi[6], G1);
    gemm_wmma_kernel<<<dim3(M_DEC / 64, 512 / 128), 256, 0, stream>>>(
        G1, f5W, sc[7], bi[7], G2, M_DEC, 512, 512, 1);
    head_kernel<<<2048, 256, 0, stream>>>(G2, f6W, sc[8], bi[8], x2b, M_DEC);

    // 5) farthest point sampling (both branches; 64 independent problems)
    fps_kernel<<<64, 256, 0, stream>>>(x1b, x2b, (float*)d_out);
    (void)in_sizes; (void)n_in; (void)out_size; (void)ws_size;
}